// GNNLayer_54657753809402
// MI455X (gfx1250) — compile-verified
//
#include <hip/hip_runtime.h>
#include <stdint.h>

#define TPB   256
#define CHUNK 256          // edges staged per async round
#define CPB   8            // chunks per block
#define EPB   (CHUNK * CPB)

// ---- CDNA5 async global->LDS staging (ASYNCcnt-tracked DMA-style copy) ----
__device__ __forceinline__ void async_ld_b32(uint32_t lds_off, const void* gbase,
                                             uint32_t byte_off) {
  // GVS mode: mem = SADDR(64b sgpr pair) + VADDR(32b) + inst_offset
  asm volatile("global_load_async_to_lds_b32 %0, %1, %2"
               :: "v"(lds_off), "v"(byte_off), "s"(gbase)
               : "memory");
}
__device__ __forceinline__ void wait_async0() {
  asm volatile("s_wait_asynccnt 0x0" ::: "memory");
}
__device__ __forceinline__ uint32_t lds_off_of(const void* p) {
  // generic "shared" address = {SHARED_BASE[63:32], lds_offset[31:0]}
  return (uint32_t)(uintptr_t)p;
}
__device__ __forceinline__ int imin(int a, int b) { return a < b ? a : b; }
__device__ __forceinline__ int imax(int a, int b) { return a > b ? a : b; }

// ---------------------------------------------------------------------------
// Kernel 1: xT[n*64+b] = x[b*N+n]  and  acc[n*64+b] = bias[n]
// LDS 64x64 tile, +1 padding => conflict-free both phases. 32-bit indexing.
// ---------------------------------------------------------------------------
__global__ void __launch_bounds__(TPB)
transpose_bias_init(const float* __restrict__ x, const float* __restrict__ bias,
                    float* __restrict__ xT, float* __restrict__ acc, int N) {
  __shared__ float tile[64][65];
  const int tx = threadIdx.x & 63;   // fast index
  const int ty = threadIdx.x >> 6;   // 0..3
  const int n0 = blockIdx.x * 64;

  // load: i = batch row b (0..63), tx = local n. coalesced along n.
  const int n_rd = n0 + tx;
  if (n_rd < N) {
    #pragma unroll
    for (int i = ty; i < 64; i += 4)
      tile[i][tx] = x[(uint32_t)i * (uint32_t)N + (uint32_t)n_rd];
  }
  __syncthreads();
  // store: i = local n, tx = b. coalesced along b.
  #pragma unroll
  for (int i = ty; i < 64; i += 4) {
    const int n = n0 + i;
    if (n < N) {
      xT[(uint32_t)n * 64u + tx]  = tile[tx][i];   // stride-65 LDS read: no conflict
      acc[(uint32_t)n * 64u + tx] = bias[n];       // bias folded into accumulator init
    }
  }
}

// ---------------------------------------------------------------------------
// Kernel 2: per-edge scatter, one wave32 per edge, float2 per lane:
//   acc[dst*64 + 2l..2l+1] += (adj[e]*w[e]) * xT[src*64 + 2l..2l+1]
// Edge metadata double-buffered into LDS via global_load_async_to_lds_b32.
// ---------------------------------------------------------------------------
__global__ void __launch_bounds__(TPB)
edge_scatter(const int* __restrict__ esrc, const int* __restrict__ edst,
             const float* __restrict__ av, const float* __restrict__ ew,
             const float* __restrict__ xT, float* __restrict__ acc, int E) {
  __shared__ int   s_src[2][CHUNK];
  __shared__ int   s_dst[2][CHUNK];
  __shared__ float s_a[2][CHUNK];
  __shared__ float s_w[2][CHUNK];

  const int t    = threadIdx.x;
  const int b2   = (t & 31) * 2;       // this lane's batch pair (0,2,...,62)
  const int wv   = t >> 5;             // wave id 0..7 -> 32 edges each per chunk
  const int base = (int)blockIdx.x * EPB;

  // stage chunk 0 (one edge's 4 fields per thread)
  {
    const int e = base + t;
    if (e < E) {
      const uint32_t off = (uint32_t)e * 4u;
      async_ld_b32(lds_off_of(&s_src[0][t]), esrc, off);
      async_ld_b32(lds_off_of(&s_dst[0][t]), edst, off);
      async_ld_b32(lds_off_of(&s_a[0][t]),   av,   off);
      async_ld_b32(lds_off_of(&s_w[0][t]),   ew,   off);
    }
  }
  wait_async0();
  __syncthreads();

  for (int c = 0; c < CPB; ++c) {
    const int cur = c & 1, nxt = cur ^ 1;

    // asynchronously prefetch next chunk's metadata (overlaps atomic phase)
    if (c + 1 < CPB) {
      const int e = base + (c + 1) * CHUNK + t;
      if (e < E) {
        const uint32_t off = (uint32_t)e * 4u;
        async_ld_b32(lds_off_of(&s_src[nxt][t]), esrc, off);
        async_ld_b32(lds_off_of(&s_dst[nxt][t]), edst, off);
        async_ld_b32(lds_off_of(&s_a[nxt][t]),   av,   off);
        async_ld_b32(lds_off_of(&s_w[nxt][t]),   ew,   off);
      }
    }

    // consume: this wave owns edges [wv*32, wv*32+32) of the chunk.
    // hoisted trip count => branch-free inner loop.
    const int chunk0 = base + c * CHUNK;
    const int kb     = wv * 32;
    const int kcnt   = imax(0, imin(32, E - chunk0 - kb));
    #pragma unroll 4
    for (int k = 0; k < kcnt; ++k) {
      const int   el  = kb + k;
      const int   src = s_src[cur][el];              // LDS broadcast reads
      const int   dst = s_dst[cur][el];
      const float cc  = s_a[cur][el] * s_w[cur][el];
      if (k + 1 < kcnt) {                            // hide data-dependent gather
        const int srcn = s_src[cur][el + 1];
        __builtin_prefetch(&xT[(uint32_t)srcn * 64u + b2], 0, 0);
      }
      const float2 g = *(const float2*)&xT[(uint32_t)src * 64u + b2]; // 256B/wave burst
      float* ap = &acc[(uint32_t)dst * 64u + b2];
      unsafeAtomicAdd(ap,     cc * g.x);             // global_atomic_add_f32 @ L2
      unsafeAtomicAdd(ap + 1, cc * g.y);
    }

    wait_async0();       // own async loads landed in LDS
    __syncthreads();     // whole block's buffer ready / safe to overwrite
  }
}

// ---------------------------------------------------------------------------
// Kernel 3: out[b*N+n] = relu(acc[n*64+b])   (tiled transpose + ReLU)
// ---------------------------------------------------------------------------
__global__ void __launch_bounds__(TPB)
finalize_relu(const float* __restrict__ acc, float* __restrict__ out, int N) {
  __shared__ float tile[64][65];
  const int tx = threadIdx.x & 63;
  const int ty = threadIdx.x >> 6;
  const int n0 = blockIdx.x * 64;

  // load: i = local n, tx = b. coalesced reads of acc.
  #pragma unroll
  for (int i = ty; i < 64; i += 4) {
    const int n = n0 + i;
    if (n < N) tile[tx][i] = acc[(uint32_t)n * 64u + tx]; // stride-65 store: no conflict
  }
  __syncthreads();
  // store: i = b, tx = local n. coalesced writes of out.
  const int n_wr = n0 + tx;
  if (n_wr < N) {
    #pragma unroll
    for (int i = ty; i < 64; i += 4) {
      const float v = tile[i][tx];                        // stride-1 read
      out[(uint32_t)i * (uint32_t)N + (uint32_t)n_wr] = v > 0.0f ? v : 0.0f;
    }
  }
}

// ---------------------------------------------------------------------------
extern "C" void kernel_launch(void* const* d_in, const int* in_sizes, int n_in,
                              void* d_out, int out_size, void* d_ws, size_t ws_size,
                              hipStream_t stream) {
  const float* x    = (const float*)d_in[0];   // [B, N]
  const float* av   = (const float*)d_in[1];   // [E]
  const float* w    = (const float*)d_in[2];   // [E]
  const float* bias = (const float*)d_in[3];   // [N]
  const int*   esrc = (const int*)d_in[4];     // [E]
  const int*   edst = (const int*)d_in[5];     // [E]

  const int N = in_sizes[3];
  const int E = in_sizes[1];

  float* xT  = (float*)d_ws;                   // [N][64]  12.8 MB
  float* acc = xT + (size_t)N * 64;            // [N][64]  12.8 MB

  const int nb = (N + 63) / 64;
  transpose_bias_init<<<nb, TPB, 0, stream>>>(x, bias, xT, acc, N);

  const int eb = (E + EPB - 1) / EPB;
  edge_scatter<<<eb, TPB, 0, stream>>>(esrc, edst, av, w, xT, acc, E);

  finalize_relu<<<nb, TPB, 0, stream>>>(acc, (float*)d_out, N);
}